// GCN_6270652252977
// MI455X (gfx1250) — compile-verified
//
#include <hip/hip_runtime.h>
#include <hip/hip_bf16.h>

typedef __attribute__((ext_vector_type(2))) float v2f;
typedef __attribute__((ext_vector_type(8))) float v8f;

#define FEAT 128

// ---------------- degree / normalization ----------------
__global__ void k_deg_init(float* deg, int n) {
    int i = blockIdx.x * blockDim.x + threadIdx.x;
    if (i < n) deg[i] = 1.0f;  // self-loop contributes 1
}

__global__ void k_deg_edges(const int* __restrict__ dst, float* deg, int E) {
    int e = blockIdx.x * blockDim.x + threadIdx.x;
    if (e < E) atomicAdd(&deg[dst[e]], 1.0f);
}

__global__ void k_dinv(float* deg, int n) {
    int i = blockIdx.x * blockDim.x + threadIdx.x;
    if (i < n) deg[i] = rsqrtf(deg[i]);  // deg >= 1 always (self-loops)
}

// ---------------- dense GEMM h1 = x @ W1 via V_WMMA_F32_16X16X4_F32 ----------------
// block = 256 threads = 8 waves; wave w computes the 16x16 tile at
// rows [blockIdx.x*16, +16), cols [w*16, +16). K-loop: 128/4 = 32 WMMA steps.
__global__ void k_gemm_wmma(const float* __restrict__ x, const float* __restrict__ W,
                            float* __restrict__ h) {
    const int lane = threadIdx.x & 31;
    const int wave = threadIdx.x >> 5;
    const int m    = lane & 15;   // row within tile (A), col within tile (B/D)
    const int half = lane >> 4;   // K sub-block selector per ISA layout
    const int rowBase = blockIdx.x * 16;
    const int colBase = wave * 16;

    const float* __restrict__ xrow = x + (size_t)(rowBase + m) * FEAT;

    v8f c = {};
#pragma unroll
    for (int k = 0; k < FEAT; k += 4) {
        const int ka = k + half * 2;
        // A 16x4 f32: lanes 0-15 -> K = k,k+1 ; lanes 16-31 -> K = k+2,k+3
        v2f a = *(const v2f*)(xrow + ka);
        // B 4x16 f32: mirrored layout, n = lane%16
        v2f b;
        b.x = W[(size_t)(ka + 0) * FEAT + colBase + m];
        b.y = W[(size_t)(ka + 1) * FEAT + colBase + m];
        c = __builtin_amdgcn_wmma_f32_16x16x4_f32(false, a, false, b, (short)0, c,
                                                  false, false);
    }
    // D 16x16 f32: lane group selects N; VGPR v holds M = v + 8*half
#pragma unroll
    for (int v = 0; v < 8; ++v) {
        const int mOut = v + 8 * half;
        h[(size_t)(rowBase + mOut) * FEAT + colBase + m] = c[v];
    }
}

// ---------------- layer-1 self-loop init: agg[i,:] = dinv[i]^2 * h1[i,:] ----------------
__global__ void k_selfloop1(const float* __restrict__ h, const float* __restrict__ dinv,
                            float* __restrict__ agg, int n32) {
    int t = blockIdx.x * blockDim.x + threadIdx.x;  // one float4 per thread
    if (t < n32) {
        int node = t >> 5;
        float di = dinv[node];
        float c  = di * di;
        float4 v = ((const float4*)h)[t];
        float4 r = make_float4(c * v.x, c * v.y, c * v.z, c * v.w);
        ((float4*)agg)[t] = r;
    }
}

// ---------------- layer-1 edge scatter: one wave per edge, float4 per lane ----------------
__global__ void k_edges1(const int* __restrict__ src, const int* __restrict__ dst,
                         const float* __restrict__ dinv, const float* __restrict__ h,
                         float* agg, int E) {
    int gid  = blockIdx.x * blockDim.x + threadIdx.x;
    int e    = gid >> 5;
    int lane = gid & 31;
    if (e < E) {
        int s = src[e], d = dst[e];
        float c = dinv[s] * dinv[d];
        float4 m = ((const float4*)(h + (size_t)s * FEAT))[lane];
        float* ad = agg + (size_t)d * FEAT + lane * 4;
        atomicAdd(ad + 0, c * m.x);
        atomicAdd(ad + 1, c * m.y);
        atomicAdd(ad + 2, c * m.z);
        atomicAdd(ad + 3, c * m.w);
    }
}

// -------- fused: a = relu(agg + b1); s2 = a @ W2; out = b2 + dinv^2 * s2 (self-loop) --------
__global__ void k_relu_dot(const float* __restrict__ agg, const float* __restrict__ b1,
                           const float* __restrict__ W2, const float* __restrict__ b2,
                           const float* __restrict__ dinv, float* __restrict__ s2,
                           float* __restrict__ out, int N) {
    int gid  = blockIdx.x * blockDim.x + threadIdx.x;
    int node = gid >> 5;
    int lane = gid & 31;
    if (node >= N) return;
    float4 v  = ((const float4*)(agg + (size_t)node * FEAT))[lane];
    float4 bb = ((const float4*)b1)[lane];
    float4 w  = ((const float4*)W2)[lane];
    float a0 = fmaxf(v.x + bb.x, 0.0f);
    float a1 = fmaxf(v.y + bb.y, 0.0f);
    float a2 = fmaxf(v.z + bb.z, 0.0f);
    float a3 = fmaxf(v.w + bb.w, 0.0f);
    float acc = a0 * w.x + a1 * w.y + a2 * w.z + a3 * w.w;
#pragma unroll
    for (int off = 16; off > 0; off >>= 1)  // wave32 reduction
        acc += __shfl_down(acc, off, 32);
    if (lane == 0) {
        s2[node] = acc;
        float di = dinv[node];
        out[node] = b2[0] + di * di * acc;  // self-loop term + bias
    }
}

// ---------------- layer-2 edge scatter (scalar per edge) ----------------
__global__ void k_edges2(const int* __restrict__ src, const int* __restrict__ dst,
                         const float* __restrict__ dinv, const float* __restrict__ s2,
                         float* out, int E) {
    int e = blockIdx.x * blockDim.x + threadIdx.x;
    if (e < E) {
        int s = src[e], d = dst[e];
        atomicAdd(&out[d], dinv[s] * dinv[d] * s2[s]);
    }
}

extern "C" void kernel_launch(void* const* d_in, const int* in_sizes, int n_in,
                              void* d_out, int out_size, void* d_ws, size_t ws_size,
                              hipStream_t stream) {
    const float* x  = (const float*)d_in[0];
    const float* W1 = (const float*)d_in[1];
    const float* b1 = (const float*)d_in[2];
    const float* W2 = (const float*)d_in[3];
    const float* b2 = (const float*)d_in[4];
    const int*   ei = (const int*)d_in[5];

    const int N = in_sizes[0] / FEAT;   // 100000
    const int E = in_sizes[5] / 2;      // 3200000
    const int* src = ei;
    const int* dst = ei + E;

    float* ws   = (float*)d_ws;
    float* dinv = ws;                            // N floats
    float* s2   = ws + N;                        // N floats
    float* h1   = ws + 2 * (size_t)N;            // N*128 floats
    float* agg  = h1 + (size_t)N * FEAT;         // N*128 floats
    float* out  = (float*)d_out;

    const int T = 256;

    // 1) degrees (with self-loop) -> dinv
    k_deg_init<<<(N + T - 1) / T, T, 0, stream>>>(dinv, N);
    k_deg_edges<<<(E + T - 1) / T, T, 0, stream>>>(dst, dinv, E);
    k_dinv<<<(N + T - 1) / T, T, 0, stream>>>(dinv, N);

    // 2) h1 = x @ W1 (WMMA f32 16x16x4); N divisible by 16, 8 col-tiles per block
    k_gemm_wmma<<<N / 16, T, 0, stream>>>(x, W1, h1);

    // 3) layer-1 aggregation
    k_selfloop1<<<((N * 32) + T - 1) / T, T, 0, stream>>>(h1, dinv, agg, N * 32);
    {
        long long threads = (long long)E * 32;
        int blocks = (int)((threads + T - 1) / T);
        k_edges1<<<blocks, T, 0, stream>>>(src, dst, dinv, h1, agg, E);
    }

    // 4) relu + bias + dot with W2, self-loop term of layer 2 into out
    k_relu_dot<<<((N * 32) + T - 1) / T, T, 0, stream>>>(agg, b1, W2, b2, dinv, s2, out, N);

    // 5) layer-2 edge aggregation
    k_edges2<<<(E + T - 1) / T, T, 0, stream>>>(src, dst, dinv, s2, out, E);
}